// GroupFeature_17678085390962
// MI455X (gfx1250) — compile-verified
//
#include <hip/hip_runtime.h>

// CDNA5 / gfx1250: wave32, WMMA 16x16x4 F32.
typedef float v2f __attribute__((ext_vector_type(2)));
typedef float v8f __attribute__((ext_vector_type(8)));

#define B_ 2
#define N_ 8192
#define C_ 128
#define K_ 16

#define FLT_BIG 3.4e38f

// ---------------------------------------------------------------------------
// Phase 0: pack xyz -> float4 {x, y, z, x^2+y^2+z^2}
// ---------------------------------------------------------------------------
__global__ void __launch_bounds__(256) pack_kernel(const float* __restrict__ xyz,
                                                   float4* __restrict__ pack,
                                                   int total) {
    int i = blockIdx.x * blockDim.x + threadIdx.x;
    if (i >= total) return;
    float x = xyz[3 * i + 0];
    float y = xyz[3 * i + 1];
    float z = xyz[3 * i + 2];
    pack[i] = make_float4(x, y, z, x * x + y * y + z * z);
}

// ---------------------------------------------------------------------------
// Phase A: fused distance + top-16 via V_WMMA_F32_16X16X4_F32.
//
// Score S[m][n] = sq[c_n] - 2*q_m.c_n  (d2 minus per-row constant sq[q_m]:
// same per-row ordering). The constant term rides in the K=3 slot:
//   A row m: (-2x, -2y, -2z, 1.0)     B col n: (x_n, y_n, z_n, sq_n)
// so D = A*B + 0 needs no epilogue at all.
//
// One wave owns 32 query rows (2 WMMA tiles per 16-candidate tile). The
// 32x16 score tile is transposed through LDS (row stride 20 dwords, b128-
// aligned). Each lane then owns one query row: it builds a 16-bit pass mask
// vs its current 16th-best and pops passes in a uniform while loop, doing a
// branch-free cmp/cndmask shift-insert into a register-resident sorted
// top-16. EXEC stays all-1s everywhere (WMMA requirement); all divergence
// is handled by selects, all branches are wave-uniform.
// ---------------------------------------------------------------------------
__global__ void __launch_bounds__(256) knn_kernel(const float4* __restrict__ pack,
                                                  int* __restrict__ idx_out) {
    __shared__ __align__(16) float dist_t[8][32][20];   // [wave][row][col]

    const int lane = threadIdx.x & 31;
    const int wave = threadIdx.x >> 5;
    const int q0   = blockIdx.x * 256 + wave * 32;   // first global query of this wave
    const int b    = q0 / N_;                        // 32 queries never straddle a batch
    const float4* __restrict__ pb = pack + (size_t)b * N_;

    const int  m      = lane & 15;
    const bool loHalf = (lane < 16);

    // A matrices (-2 folded in; K3 column = 1.0 carried by the upper half-wave).
    float4 pq0 = pack[q0 + m];
    float4 pq1 = pack[q0 + 16 + m];
    v2f a0, a1;
    a0.x = loHalf ? (-2.0f * pq0.x) : (-2.0f * pq0.z);
    a0.y = loHalf ? (-2.0f * pq0.y) : 1.0f;
    a1.x = loHalf ? (-2.0f * pq1.x) : (-2.0f * pq1.z);
    a1.y = loHalf ? (-2.0f * pq1.y) : 1.0f;

    const v8f czero = {};   // loop-invariant zero accumulator

    // Sorted (ascending) top-16 for query row q0+lane.
    float bd[K_];
    int   bi[K_];
#pragma unroll
    for (int t = 0; t < K_; ++t) { bd[t] = FLT_BIG; bi[t] = 0; }

    const int rb = loHalf ? 0 : 8;   // WMMA C/D row base for this lane

    for (int ct = 0; ct < N_ / 16; ++ct) {
        // Candidate tile: lane owns column n = lane&15.
        float4 pc = pb[ct * 16 + m];
        v2f bm;
        bm.x = loHalf ? pc.x : pc.z;
        bm.y = loHalf ? pc.y : pc.w;   // K3 row carries sq[c_n]

        v8f d0 = __builtin_amdgcn_wmma_f32_16x16x4_f32(
            false, a0, false, bm, (short)0, czero, false, false);
        v8f d1 = __builtin_amdgcn_wmma_f32_16x16x4_f32(
            false, a1, false, bm, (short)0, czero, false, false);

        // Transpose 32x16 score tile through LDS.
#pragma unroll
        for (int r = 0; r < 8; ++r) dist_t[wave][rb + r][m] = d0[r];
#pragma unroll
        for (int r = 0; r < 8; ++r) dist_t[wave][16 + rb + r][m] = d1[r];

        __builtin_amdgcn_wave_barrier();   // same-wave LDS is in-order; pin schedule

        // Each lane reads its own row (4x ds_load_b128, conflict-light).
        const float4 r0 = *(const float4*)&dist_t[wave][lane][0];
        const float4 r1 = *(const float4*)&dist_t[wave][lane][4];
        const float4 r2 = *(const float4*)&dist_t[wave][lane][8];
        const float4 r3 = *(const float4*)&dist_t[wave][lane][12];

        // 16-bit pass mask vs current threshold (stale within tile is fine:
        // the insert below is exact and no-ops for non-qualifying values).
        const float th = bd[K_ - 1];
        unsigned mask = 0;
        mask |= (r0.x < th) ? 0x0001u : 0u;
        mask |= (r0.y < th) ? 0x0002u : 0u;
        mask |= (r0.z < th) ? 0x0004u : 0u;
        mask |= (r0.w < th) ? 0x0008u : 0u;
        mask |= (r1.x < th) ? 0x0010u : 0u;
        mask |= (r1.y < th) ? 0x0020u : 0u;
        mask |= (r1.z < th) ? 0x0040u : 0u;
        mask |= (r1.w < th) ? 0x0080u : 0u;
        mask |= (r2.x < th) ? 0x0100u : 0u;
        mask |= (r2.y < th) ? 0x0200u : 0u;
        mask |= (r2.z < th) ? 0x0400u : 0u;
        mask |= (r2.w < th) ? 0x0800u : 0u;
        mask |= (r3.x < th) ? 0x1000u : 0u;
        mask |= (r3.y < th) ? 0x2000u : 0u;
        mask |= (r3.z < th) ? 0x4000u : 0u;
        mask |= (r3.w < th) ? 0x8000u : 0u;

        const int cbase = ct * 16;
        // Uniform flush loop: one candidate per lane per iteration.
        while (__any((int)(mask != 0u))) {
            const bool act = (mask != 0u);
            const int  j   = __builtin_ctz(mask | 0x10000u);   // 16 when empty (safe slot)
            mask &= mask - 1u;

            float dv = dist_t[wave][lane][j];      // dynamic LDS read
            dv = act ? dv : FLT_BIG;               // inactive lanes -> no-op insert
            const int cid = cbase + j;

            // Branch-free shift-insert, fixed register indices only.
            bool c[K_];
#pragma unroll
            for (int t = 0; t < K_; ++t) c[t] = dv < bd[t];
            float nb[K_];
            int   ni[K_];
            nb[0] = c[0] ? dv : bd[0];
            ni[0] = c[0] ? cid : bi[0];
#pragma unroll
            for (int t = 1; t < K_; ++t) {
                nb[t] = c[t] ? (c[t - 1] ? bd[t - 1] : dv) : bd[t];
                ni[t] = c[t] ? (c[t - 1] ? bi[t - 1] : cid) : bi[t];
            }
#pragma unroll
            for (int t = 0; t < K_; ++t) { bd[t] = nb[t]; bi[t] = ni[t]; }
        }

        __builtin_amdgcn_wave_barrier();   // scores consumed before next tile's stores
    }

    int* __restrict__ op = idx_out + (size_t)(q0 + lane) * K_;
#pragma unroll
    for (int t = 0; t < K_; ++t) op[t] = bi[t];
}

// ---------------------------------------------------------------------------
// Phase B: gather neighborhoods (bandwidth-bound, ~270 MB total traffic).
// One block per query; float4-vectorized feature streaming.
// ---------------------------------------------------------------------------
__global__ void __launch_bounds__(128) gather_kernel(const float4* __restrict__ pack,
                                                     const float* __restrict__ feat,
                                                     const int* __restrict__ idx,
                                                     float* __restrict__ outN,
                                                     float* __restrict__ outF) {
    const int q = blockIdx.x;       // global query index in [0, B*N)
    const int b = q / N_;
    const int t = threadIdx.x;

    __shared__ int    sidx[K_];
    __shared__ float4 ctr;
    if (t < K_)  sidx[t] = idx[(size_t)q * K_ + t];
    if (t == 0)  ctr = pack[q];
    __syncthreads();

    // Features as float4 rows of 32: thread (k0, l4) covers k = k0 + 4*kk.
    const float4* __restrict__ fb4 = (const float4*)(feat + (size_t)b * N_ * C_);
    float4* __restrict__ oF4 = (float4*)(outF + (size_t)q * K_ * C_);
    const int k0 = t >> 5;          // 0..3
    const int l4 = t & 31;          // 0..31
#pragma unroll
    for (int kk = 0; kk < 4; ++kk) {
        const int k = k0 + kk * 4;
        oF4[(size_t)k * 32 + l4] = fb4[(size_t)sidx[k] * 32 + l4];
    }

    // Center-subtracted coordinates.
    if (t < K_ * 3) {
        const int k = t / 3, d = t % 3;
        const float4* __restrict__ pb = pack + (size_t)b * N_;
        float4 p = pb[sidx[k]];
        float pv = (d == 0) ? p.x   : ((d == 1) ? p.y   : p.z);
        float cv = (d == 0) ? ctr.x : ((d == 1) ? ctr.y : ctr.z);
        outN[((size_t)q * K_ + k) * 3 + d] = pv - cv;
    }
}

// ---------------------------------------------------------------------------
extern "C" void kernel_launch(void* const* d_in, const int* in_sizes, int n_in,
                              void* d_out, int out_size, void* d_ws, size_t ws_size,
                              hipStream_t stream) {
    const float* xyz  = (const float*)d_in[0];   // [B, N, 3] f32
    const float* feat = (const float*)d_in[1];   // [B, N, C] f32
    // d_in[2] is group_size (==16), baked in at compile time.

    // Workspace layout: [idx: B*N*K ints][pack: B*N float4]
    int*    idx  = (int*)d_ws;
    float4* pack = (float4*)((char*)d_ws + (size_t)B_ * N_ * K_ * sizeof(int));

    float* outN = (float*)d_out;                          // [B, N, K, 3]
    float* outF = outN + (size_t)B_ * N_ * K_ * 3;        // [B, N, K, C]

    const int total = B_ * N_;
    pack_kernel<<<(total + 255) / 256, 256, 0, stream>>>(xyz, pack, total);
    knn_kernel<<<total / 256, 256, 0, stream>>>(pack, idx);
    gather_kernel<<<total, 128, 0, stream>>>(pack, feat, idx, outN, outF);
}